// MyGIN_36344013259383
// MI455X (gfx1250) — compile-verified
//
#include <hip/hip_runtime.h>
#include <hip/hip_bf16.h>

typedef __attribute__((ext_vector_type(2))) float v2f;
typedef __attribute__((ext_vector_type(8))) float v8f;

// ---------------------------------------------------------------------------
// Kernel 1: edge message + scatter-add.  agg[dst] += relu(x[src])
// One thread per (edge, 4-float chunk). dim fixed = 64 (16 chunks/edge).
// Pure bandwidth + global_atomic_add_f32; this is the roofline-dominant stage.
// ---------------------------------------------------------------------------
__global__ __launch_bounds__(256)
void gin_scatter_relu_add(const float* __restrict__ x,
                          const long long* __restrict__ src,
                          const long long* __restrict__ dst,
                          float* __restrict__ agg,
                          int E)
{
    const int CH = 16;                      // 64 floats / 4
    long long tid = (long long)blockIdx.x * blockDim.x + threadIdx.x;
    long long total = (long long)E * CH;
    if (tid >= total) return;
    long long e = tid / CH;
    int c = (int)(tid % CH) * 4;
    long long s = src[e];
    long long d = dst[e];
    const float4* xp = (const float4*)(x + s * 64 + c);
    float4 v = *xp;
    v.x = fmaxf(v.x, 0.f); v.y = fmaxf(v.y, 0.f);
    v.z = fmaxf(v.z, 0.f); v.w = fmaxf(v.w, 0.f);
    float* ap = agg + d * 64 + c;
    atomicAdd(ap + 0, v.x);
    atomicAdd(ap + 1, v.y);
    atomicAdd(ap + 2, v.z);
    atomicAdd(ap + 3, v.w);
}

// ---------------------------------------------------------------------------
// Kernel 2: GIN node MLP with fp32 WMMA (V_WMMA_F32_16X16X4_F32).
//   hpre = (1+eps)*x + agg                    [16-row tile per wave, in LDS]
//   t    = relu(hpre @ W1 + b1)               [GEMM1: K1 x N1]
//   out  = (t @ W2 + b2)  (+optional relu)    [GEMM2: N1 x N2]
// W1/W2 are zero-padded to (K1 x N1) / (N1 x N2) while staging into LDS.
// Fragment layouts per CDNA5 ISA 7.12.2 (wave32):
//   A 16x4 f32 : 2 VGPRs; lanes 0-15 -> K={0,1}, lanes 16-31 -> K={2,3}
//   B  4x16 f32: 2 VGPRs; lanes 0-15 -> K rows {0,1}, lanes 16-31 -> {2,3}
//   C/D 16x16  : 8 VGPRs; lanes 0-15 -> M=r, lanes 16-31 -> M=r+8
// ---------------------------------------------------------------------------
template<int K1, int N1, int N2, bool RELU_OUT>
__global__ __launch_bounds__(256)
void gin_mlp_wmma(const float* __restrict__ xin,   // nrows x K1
                  const float* __restrict__ agg,   // nrows x K1
                  const float* __restrict__ eps,   // [1]
                  const float* __restrict__ W1, int w1cols,  // K1 x w1cols (valid part)
                  const float* __restrict__ b1, int b1n,
                  const float* __restrict__ W2, int w2rows, int w2cols,
                  const float* __restrict__ b2, int b2n,
                  float* __restrict__ out, int outN,         // nrows x outN (outN == N2)
                  int nrows)
{
    constexpr int WAVES = 8;
    __shared__ float sW1[K1 * N1];
    __shared__ float sW2[N1 * N2];
    __shared__ float sT[WAVES][16 * K1];   // K1 >= N1 for all instantiations

    const int tid  = threadIdx.x;
    const int wave = tid >> 5;
    const int lane = tid & 31;
    const int laneLo = lane & 15;
    const int hiHalf = lane >> 4;          // 0 for lanes 0-15, 1 for 16-31

    // ---- stage (zero-padded) weights into LDS, whole block cooperates ----
    for (int i = tid; i < K1 * N1; i += blockDim.x) {
        int r = i / N1, c = i % N1;
        sW1[i] = (c < w1cols) ? W1[r * w1cols + c] : 0.f;
    }
    for (int i = tid; i < N1 * N2; i += blockDim.x) {
        int r = i / N2, c = i % N2;
        sW2[i] = (r < w2rows && c < w2cols) ? W2[r * w2cols + c] : 0.f;
    }
    __syncthreads();

    const float onepe = 1.0f + eps[0];
    const int rowBase = (blockIdx.x * WAVES + wave) * 16;

    // ---- stage hpre tile (16 x K1) into this wave's LDS slab ----
    for (int i = lane; i < 16 * K1; i += 32) {
        int r = i / K1, c = i % K1;
        int g = rowBase + r;
        float v = 0.f;
        if (g < nrows) {
            size_t o = (size_t)g * K1 + c;
            v = onepe * xin[o] + agg[o];
        }
        sT[wave][i] = v;
    }
    // same-wave DS ops are in-order; no block barrier needed for private slab

    // ---- GEMM1: t = relu(hpre @ W1 + b1), accumulate N1/16 tiles ----
    v8f acc1[N1 / 16];
#pragma unroll
    for (int nt = 0; nt < N1 / 16; ++nt) acc1[nt] = {};

#pragma unroll
    for (int s = 0; s < K1 / 4; ++s) {
        int kb = 4 * s + (hiHalf << 1);
        v2f a;
        a.x = sT[wave][laneLo * K1 + kb + 0];
        a.y = sT[wave][laneLo * K1 + kb + 1];
#pragma unroll
        for (int nt = 0; nt < N1 / 16; ++nt) {
            int col = nt * 16 + laneLo;
            v2f b;
            b.x = sW1[(kb + 0) * N1 + col];
            b.y = sW1[(kb + 1) * N1 + col];
            acc1[nt] = __builtin_amdgcn_wmma_f32_16x16x4_f32(
                false, a, false, b, (short)0, acc1[nt], false, false);
        }
    }

    // ---- relu + bias, write t (16 x N1) back into the same LDS slab ----
#pragma unroll
    for (int nt = 0; nt < N1 / 16; ++nt) {
        int col = nt * 16 + laneLo;
        float bias = (col < b1n) ? b1[col] : 0.f;
#pragma unroll
        for (int r = 0; r < 8; ++r) {
            int row = r + (hiHalf << 3);
            float v = acc1[nt][r] + bias;
            v = fmaxf(v, 0.f);
            sT[wave][row * N1 + col] = v;
        }
    }

    // ---- GEMM2: out = t @ W2 + b2 ----
    v8f acc2[N2 / 16];
#pragma unroll
    for (int nt = 0; nt < N2 / 16; ++nt) acc2[nt] = {};

#pragma unroll
    for (int s = 0; s < N1 / 4; ++s) {
        int kb = 4 * s + (hiHalf << 1);
        v2f a;
        a.x = sT[wave][laneLo * N1 + kb + 0];
        a.y = sT[wave][laneLo * N1 + kb + 1];
#pragma unroll
        for (int nt = 0; nt < N2 / 16; ++nt) {
            int col = nt * 16 + laneLo;
            v2f b;
            b.x = sW2[(kb + 0) * N2 + col];
            b.y = sW2[(kb + 1) * N2 + col];
            acc2[nt] = __builtin_amdgcn_wmma_f32_16x16x4_f32(
                false, a, false, b, (short)0, acc2[nt], false, false);
        }
    }

    // ---- epilogue: bias (+optional relu), store to global ----
#pragma unroll
    for (int nt = 0; nt < N2 / 16; ++nt) {
        int col = nt * 16 + laneLo;
        float bias = (col < b2n) ? b2[col] : 0.f;
#pragma unroll
        for (int r = 0; r < 8; ++r) {
            int row = r + (hiHalf << 3);
            float v = acc2[nt][r] + bias;
            if (RELU_OUT) v = fmaxf(v, 0.f);
            int g = rowBase + row;
            if (g < nrows) out[(size_t)g * outN + col] = v;
        }
    }
}

// ---------------------------------------------------------------------------
// Kernel 3: row softmax over 40 valid columns (input stride 48, padded).
// One wave32 per row; lane covers cols {lane, lane+32}.
// ---------------------------------------------------------------------------
__global__ __launch_bounds__(256)
void gin_softmax40(const float* __restrict__ in, int inStride,
                   float* __restrict__ out, int nrows)
{
    const int lane = threadIdx.x & 31;
    const int rowsPerBlock = blockDim.x >> 5;
    const int row = blockIdx.x * rowsPerBlock + (threadIdx.x >> 5);
    if (row >= nrows) return;

    const float* p = in + (size_t)row * inStride;
    float v0 = (lane      < 40) ? p[lane]      : -3.4e38f;
    float v1 = (lane + 32 < 40) ? p[lane + 32] : -3.4e38f;

    float m = fmaxf(v0, v1);
#pragma unroll
    for (int off = 16; off > 0; off >>= 1)
        m = fmaxf(m, __shfl_xor(m, off, 32));

    float e0 = (lane      < 40) ? __expf(v0 - m) : 0.f;
    float e1 = (lane + 32 < 40) ? __expf(v1 - m) : 0.f;
    float s = e0 + e1;
#pragma unroll
    for (int off = 16; off > 0; off >>= 1)
        s += __shfl_xor(s, off, 32);

    float inv = 1.0f / s;
    if (lane      < 40) out[(size_t)row * 40 + lane]      = e0 * inv;
    if (lane + 32 < 40) out[(size_t)row * 40 + lane + 32] = e1 * inv;
}

// ---------------------------------------------------------------------------
// Host-side launch sequence (graph-capture safe: only kernels + memsetAsync).
// ---------------------------------------------------------------------------
extern "C" void kernel_launch(void* const* d_in, const int* in_sizes, int n_in,
                              void* d_out, int out_size, void* d_ws, size_t ws_size,
                              hipStream_t stream)
{
    const float*     x    = (const float*)d_in[0];
    const long long* eidx = (const long long*)d_in[1];
    const float*     W1a  = (const float*)d_in[2];
    const float*     b1a  = (const float*)d_in[3];
    const float*     W2a  = (const float*)d_in[4];
    const float*     b2a  = (const float*)d_in[5];
    const float*     epsa = (const float*)d_in[6];
    const float*     W1b  = (const float*)d_in[7];
    const float*     b1b  = (const float*)d_in[8];
    const float*     W2b  = (const float*)d_in[9];
    const float*     b2b  = (const float*)d_in[10];
    const float*     epsb = (const float*)d_in[11];

    const int N = in_sizes[0] / 64;     // 100000
    const int E = in_sizes[1] / 2;      // 1000000
    const long long* src = eidx;
    const long long* dst = eidx + E;

    // workspace layout
    float* agg = (float*)d_ws;                       // N x 64
    float* h   = agg + (size_t)N * 64;               // N x 64
    float* hB  = h   + (size_t)N * 64;               // N x 48 (padded pre-softmax)

    const int scatterBlocks = (int)(((long long)E * 16 + 255) / 256);
    const int mlpBlocks     = (N + 127) / 128;       // 8 waves x 16 rows / block
    const int smBlocks      = (N + 7) / 8;           // 8 rows (waves) / block

    // ---- layer A ----
    hipMemsetAsync(agg, 0, (size_t)N * 64 * sizeof(float), stream);
    gin_scatter_relu_add<<<scatterBlocks, 256, 0, stream>>>(x, src, dst, agg, E);
    gin_mlp_wmma<64, 64, 64, true><<<mlpBlocks, 256, 0, stream>>>(
        x, agg, epsa, W1a, 64, b1a, 64, W2a, 64, 64, b2a, 64, h, 64, N);

    // ---- layer B (dims padded 40 -> 48) ----
    hipMemsetAsync(agg, 0, (size_t)N * 64 * sizeof(float), stream);
    gin_scatter_relu_add<<<scatterBlocks, 256, 0, stream>>>(h, src, dst, agg, E);
    gin_mlp_wmma<64, 48, 48, false><<<mlpBlocks, 256, 0, stream>>>(
        h, agg, epsb, W1b, 40, b1b, 40, W2b, 40, 40, b2b, 40, hB, 48, N);

    // ---- softmax ----
    gin_softmax40<<<smBlocks, 256, 0, stream>>>(hB, 48, (float*)d_out, N);
}